// FSEncoderClasification_29652454212331
// MI455X (gfx1250) — compile-verified
//
#include <hip/hip_runtime.h>
#include <hip/hip_bf16.h>

typedef __attribute__((ext_vector_type(2))) float v2f;
typedef __attribute__((ext_vector_type(8))) float v8f;

#define BATCH  16
#define IN_C   3
#define DIM    256
#define SET_N  128
#define NP     20
#define NCHUNK 64          // column chunk so LDS h1 tile = 256*64*4 = 64 KB
#define ZSTR   260         // padded row stride for head LDS (260 % 64 == 4 -> conflict free)

// ---------------------------------------------------------------------------
// Kernel 1: per-batch  h2 = conv2_w @ relu(conv1_w @ x + b1) + b2
// grid = 16 blocks (one batch each), 256 threads = 8 waves
// ---------------------------------------------------------------------------
__global__ __launch_bounds__(256) void k_encoder(const float* __restrict__ x,
                                                 const float* __restrict__ w1,
                                                 const float* __restrict__ b1,
                                                 const float* __restrict__ w2,
                                                 const float* __restrict__ b2,
                                                 float* __restrict__ h2out)
{
    __shared__ float h1[DIM * NCHUNK];   // 64 KB: conv1 activations, [d][n] chunk

    const int b    = blockIdx.x;
    const int tid  = threadIdx.x;
    const int lane = tid & 31;
    const int wv   = tid >> 5;
    const int lrow = lane & 15;
    const int koff = (lane < 16) ? 0 : 2;
    const float* xb = x + (size_t)b * IN_C * SET_N;

    for (int chunk = 0; chunk < SET_N / NCHUNK; ++chunk) {
        const int nbase = chunk * NCHUNK;

        // ---- conv1 + ReLU into LDS (coalesced in n across lanes) ----------
        {
            const int n     = tid & 63;
            const int dbase = tid >> 6;          // uniform per wave -> scalar w1/b1 loads
            const float x0 = xb[0 * SET_N + nbase + n];
            const float x1 = xb[1 * SET_N + nbase + n];
            const float x2 = xb[2 * SET_N + nbase + n];
            for (int i = 0; i < 64; ++i) {
                const int d = dbase + 4 * i;
                float v = w1[d * 3 + 0] * x0 + w1[d * 3 + 1] * x1 +
                          w1[d * 3 + 2] * x2 + b1[d];
                h1[d * NCHUNK + n] = fmaxf(v, 0.0f);
            }
        }
        __syncthreads();

        // ---- conv2 via V_WMMA_F32_16X16X4_F32 -----------------------------
        // wave wv owns M-tiles {2wv, 2wv+1}; 4 N-tiles of this chunk
        const int m0 = wv * 32;
        v8f acc[2][4];
        const v8f vzero = {};
        for (int mt = 0; mt < 2; ++mt)
            for (int nt = 0; nt < 4; ++nt) acc[mt][nt] = vzero;

        for (int k = 0; k < DIM; k += 4) {
            v2f a0, a1;
            const float* wp0 = w2 + (size_t)(m0 + lrow) * DIM + k + koff;
            const float* wp1 = w2 + (size_t)(m0 + 16 + lrow) * DIM + k + koff;
            a0.x = wp0[0]; a0.y = wp0[1];
            a1.x = wp1[0]; a1.y = wp1[1];
            for (int nt = 0; nt < 4; ++nt) {
                v2f bf;
                const float* hp = &h1[(k + koff) * NCHUNK + nt * 16 + lrow];
                bf.x = hp[0];
                bf.y = hp[NCHUNK];
                acc[0][nt] = __builtin_amdgcn_wmma_f32_16x16x4_f32(
                    false, a0, false, bf, (short)0, acc[0][nt], false, false);
                acc[1][nt] = __builtin_amdgcn_wmma_f32_16x16x4_f32(
                    false, a1, false, bf, (short)0, acc[1][nt], false, false);
            }
        }

        // ---- bias + store h2 chunk ---------------------------------------
        const int mhalf = (lane < 16) ? 0 : 8;
        for (int mt = 0; mt < 2; ++mt) {
            const int mrow0 = m0 + mt * 16 + mhalf;
            for (int nt = 0; nt < 4; ++nt) {
                const int ncol = nbase + nt * 16 + lrow;
                for (int v = 0; v < 8; ++v) {
                    const int m = mrow0 + v;
                    h2out[(size_t)b * DIM * SET_N + (size_t)m * SET_N + ncol] =
                        acc[mt][nt][v] + b2[m];
                }
            }
        }
        __syncthreads();   // before overwriting h1 with next chunk
    }
}

// ---------------------------------------------------------------------------
// Kernel 2: FSPool. One wave per (b,c) row of length 128.
// grid = 4096/8 = 512 blocks, 256 threads = 8 waves
// ---------------------------------------------------------------------------
__global__ __launch_bounds__(256) void k_fspool(const float* __restrict__ h2,
                                                const float* __restrict__ pool_w,
                                                float* __restrict__ pooled)
{
    __shared__ float srow[8][SET_N];
    const int lane = threadIdx.x & 31;
    const int wv   = threadIdx.x >> 5;
    const int row  = blockIdx.x * 8 + wv;    // 0..4095
    const int c    = row & (DIM - 1);

    float s[4];
    for (int q = 0; q < 4; ++q) {
        s[q] = h2[(size_t)row * SET_N + lane + 32 * q];
        srow[wv][lane + 32 * q] = s[q];      // wave-private region; same-wave DS is in order
    }

    // Bsum_i = sum_j |s_i - s_j|
    float bsum[4] = {0.f, 0.f, 0.f, 0.f};
    for (int j = 0; j < SET_N; ++j) {
        const float sj = srow[wv][j];
        bsum[0] += fabsf(s[0] - sj);
        bsum[1] += fabsf(s[1] - sj);
        bsum[2] += fabsf(s[2] - sj);
        bsum[3] += fabsf(s[3] - sj);
    }

    float pacc = 0.0f;
    for (int j = 0; j < SET_N; ++j) {
        const float scal = (float)(SET_N - 1 - 2 * j);   // 127 - 2j
        float l0 = scal * s[0] - bsum[0];
        float l1 = scal * s[1] - bsum[1];
        float l2 = scal * s[2] - bsum[2];
        float l3 = scal * s[3] - bsum[3];
        float m = fmaxf(fmaxf(l0, l1), fmaxf(l2, l3));
        for (int off = 16; off; off >>= 1) m = fmaxf(m, __shfl_xor(m, off, 32));
        const float e0 = __expf(l0 - m), e1 = __expf(l1 - m);
        const float e2 = __expf(l2 - m), e3 = __expf(l3 - m);
        float se = e0 + e1 + e2 + e3;
        float ss = e0 * s[0] + e1 * s[1] + e2 * s[2] + e3 * s[3];
        for (int off = 16; off; off >>= 1) {
            se += __shfl_xor(se, off, 32);
            ss += __shfl_xor(ss, off, 32);
        }
        const float xs = ss / se;

        // piecewise-linear pool weight, computed exactly as the reference
        const float ratio = (float)j / 127.0f;
        const float idxf  = (float)NP * ratio;
        int   idx  = (int)floorf(idxf);
        float frac = idxf - (float)idx;
        int   idx1 = (idx + 1 > NP) ? NP : (idx + 1);
        const float wl = pool_w[c * (NP + 1) + idx];
        const float wr = pool_w[c * (NP + 1) + idx1];
        pacc += xs * ((1.0f - frac) * wl + frac * wr);
    }
    if (lane == 0) pooled[row] = pacc;
}

// ---------------------------------------------------------------------------
// Kernel 3: head (lin1+relu, lin2, cls1+relu, cls2). Single block, 8 waves.
// z in LDS with padded stride 260 (bank-conflict free A-fragments).
// ---------------------------------------------------------------------------
__device__ inline void dense_layer(const float* __restrict__ src, float* __restrict__ dst,
                                   const float* __restrict__ W, const float* __restrict__ bias,
                                   bool do_relu, int lane, int wv)
{
    const int lrow = lane & 15;
    const int koff = (lane < 16) ? 0 : 2;
    for (int t = 0; t < 2; ++t) {                 // each wave: N-tiles wv and wv+8
        const int n0 = (wv + t * 8) * 16;
        v8f acc = {};
        for (int k = 0; k < DIM; k += 4) {
            v2f a, bf;
            a.x = src[lrow * ZSTR + k + koff];
            a.y = src[lrow * ZSTR + k + koff + 1];
            const float* wp = W + (size_t)(n0 + lrow) * DIM + k + koff;
            bf.x = wp[0]; bf.y = wp[1];
            acc = __builtin_amdgcn_wmma_f32_16x16x4_f32(
                false, a, false, bf, (short)0, acc, false, false);
        }
        const int   n  = n0 + lrow;
        const float bv = bias[n];
        const int   mb = (lane < 16) ? 0 : 8;
        for (int v = 0; v < 8; ++v) {
            float val = acc[v] + bv;
            if (do_relu) val = fmaxf(val, 0.0f);
            dst[(mb + v) * ZSTR + n] = val;
        }
    }
}

__global__ __launch_bounds__(256) void k_head(const float* __restrict__ pooled,
                                              const float* __restrict__ w_l1, const float* __restrict__ b_l1,
                                              const float* __restrict__ w_l2, const float* __restrict__ b_l2,
                                              const float* __restrict__ w_c1, const float* __restrict__ b_c1,
                                              const float* __restrict__ w_c2, const float* __restrict__ b_c2,
                                              float* __restrict__ out)
{
    __shared__ float zA[16 * ZSTR];
    __shared__ float zB[16 * ZSTR];
    const int tid  = threadIdx.x;
    const int lane = tid & 31;
    const int wv   = tid >> 5;

    for (int i = tid; i < BATCH * DIM; i += 256)
        zA[(i >> 8) * ZSTR + (i & 255)] = pooled[i];
    __syncthreads();

    dense_layer(zA, zB, w_l1, b_l1, true,  lane, wv);  __syncthreads();
    dense_layer(zB, zA, w_l2, b_l2, false, lane, wv);  __syncthreads();
    dense_layer(zA, zB, w_c1, b_c1, true,  lane, wv);  __syncthreads();

    // cls2: 16x10 output -> one 16x16 WMMA tile on wave 0, cols >= 10 zeroed
    if (wv == 0) {
        const int lrow = lane & 15;
        const int koff = (lane < 16) ? 0 : 2;
        v8f acc = {};
        for (int k = 0; k < DIM; k += 4) {
            v2f a, bf;
            a.x = zB[lrow * ZSTR + k + koff];
            a.y = zB[lrow * ZSTR + k + koff + 1];
            bf.x = 0.0f; bf.y = 0.0f;
            if (lrow < 10) {                      // avoid OOB reads of (10,256) weight
                const float* wp = w_c2 + (size_t)lrow * DIM + k + koff;
                bf.x = wp[0]; bf.y = wp[1];
            }
            acc = __builtin_amdgcn_wmma_f32_16x16x4_f32(
                false, a, false, bf, (short)0, acc, false, false);
        }
        if (lrow < 10) {
            const float bv = b_c2[lrow];
            const int   mb = (lane < 16) ? 0 : 8;
            for (int v = 0; v < 8; ++v)
                out[(mb + v) * 10 + lrow] = acc[v] + bv;
        }
    }
}

// ---------------------------------------------------------------------------
extern "C" void kernel_launch(void* const* d_in, const int* in_sizes, int n_in,
                              void* d_out, int out_size, void* d_ws, size_t ws_size,
                              hipStream_t stream)
{
    (void)in_sizes; (void)n_in; (void)out_size; (void)ws_size;
    const float* x    = (const float*)d_in[0];
    const float* w1   = (const float*)d_in[1];
    const float* b1   = (const float*)d_in[2];
    const float* w2   = (const float*)d_in[3];
    const float* b2   = (const float*)d_in[4];
    const float* pw   = (const float*)d_in[5];
    const float* wl1  = (const float*)d_in[6];
    const float* bl1  = (const float*)d_in[7];
    const float* wl2  = (const float*)d_in[8];
    const float* bl2  = (const float*)d_in[9];
    const float* wc1  = (const float*)d_in[10];
    const float* bc1  = (const float*)d_in[11];
    const float* wc2  = (const float*)d_in[12];
    const float* bc2  = (const float*)d_in[13];

    float* ws_f   = (float*)d_ws;
    float* h2     = ws_f;                                   // 16*256*128 floats (2 MB)
    float* pooled = ws_f + (size_t)BATCH * DIM * SET_N;     // 4096 floats

    k_encoder<<<BATCH, 256, 0, stream>>>(x, w1, b1, w2, b2, h2);
    k_fspool<<<(BATCH * DIM) / 8, 256, 0, stream>>>(h2, pw, pooled);
    k_head<<<1, 256, 0, stream>>>(pooled, wl1, bl1, wl2, bl2, wc1, bc1, wc2, bc2,
                                  (float*)d_out);
}